// GVPAttentionLayer_8495445311730
// MI455X (gfx1250) — compile-verified
//
#include <hip/hip_runtime.h>
#include <hip/hip_bf16.h>

#define N_  1024
#define C_  5
#define SI_ 128
#define VI_ 16
#define SE_ 32
#define SO_ 128
#define VO_ 16
#define E_  16384
#define H_  8
#define D_  16
#define L_  5

typedef __attribute__((ext_vector_type(2))) float v2f;
typedef __attribute__((ext_vector_type(8))) float v8f;

__device__ __forceinline__ float sigm(float x) { return 1.0f / (1.0f + __expf(-x)); }

// Pack weight matrix W [Ncols x Kreal] (row-major, per-output-row) into the
// WMMA B-fragment friendly layout WP[k/2][n][2] with zero K-padding to Kpad:
//   WP[p*2*Ncols + n*2 + q] = (2p+q < Kreal) ? W[n*Kreal + 2p+q] : 0
// so a lane's (ka, ka+1) pair for column n is one contiguous 8B load.
__global__ void pack_kernel(const float* __restrict__ W, float* __restrict__ WP,
                            int Kreal, int Kpad, int Ncols)
{
  int i = blockIdx.x * blockDim.x + threadIdx.x;
  int total = Kpad * Ncols;
  if (i >= total) return;
  int q = i & 1;
  int n = (i >> 1) % Ncols;
  int p = i / (2 * Ncols);
  int k = 2 * p + q;
  WP[i] = (k < Kreal) ? W[n * Kreal + k] : 0.0f;
}

// One wave computes a 16(M) x NCOLS(N) tile: D = A(16xKPAD) * WP + bias using
// V_WMMA_F32_16X16X4_F32.  A is an LDS panel (row stride Alds, K zero-padded),
// WP is the packed/padded weight panel from pack_kernel.
template <int KPAD, int NCOLS>
__device__ __forceinline__ void gemm_wmma(const float* A, int Alds,
                                          const float* __restrict__ WP,
                                          const float* __restrict__ bias,
                                          float* out, int Old, int lane)
{
  const int half = lane >> 4;    // 0: lanes 0-15, 1: lanes 16-31
  const int mn   = lane & 15;    // M for A-frag, N for B/D-frag
  const int kh   = half * 2;     // K interleave per ISA layout (0,1 | 2,3)
  for (int n0 = 0; n0 < NCOLS; n0 += 16) {
    v8f acc = {};
    const float* wbase = WP + (n0 + mn) * 2 + kh * NCOLS;
    __builtin_prefetch(wbase, 0, 3);
    for (int k0 = 0; k0 < KPAD; k0 += 4) {
      const int ka = k0 + kh;
      v2f a = *(const v2f*)(A + mn * Alds + ka);              // ds_load_b64
      v2f b = *(const v2f*)(WP + (size_t)(ka >> 1) * (2 * NCOLS) + (n0 + mn) * 2);
      acc = __builtin_amdgcn_wmma_f32_16x16x4_f32(
          false, a, false, b, (short)0, acc, false, false);
    }
    const float bb = bias[n0 + mn];
#pragma unroll
    for (int j = 0; j < 8; ++j)
      out[(half * 8 + j) * Old + (n0 + mn)] = acc[j] + bb;
  }
}

// One GVP stage (after the A-panel scalar prefix has been written):
//   vh = v^T @ wh^T ; vn ; so = [prefix, vn] @ ws^T + b (WMMA)
//   gate = act(so) @ wsv^T + b (WMMA) ; vo = (vh @ wv^T) * sigmoid(gate)
template <int NVIN, int NH, int VNCOL, int KPAD, bool GSIG>
__device__ void gvp_tail(float* A, float* VH, float* SOb, float* GT, float* VOb,
                         const float* Vsrc, int vstride,
                         const float* wh, const float* wswP, const float* wsb,
                         const float* wv, const float* wsvP, const float* wsvb,
                         int lane)
{
  // vh[row][d][h] = sum_k Vsrc[row][k][d] * wh[h][k]
  for (int i = lane; i < 16 * 3 * NH; i += 32) {
    int row = i / (3 * NH), j = i % (3 * NH);
    int d = j / NH, h = j % NH;
    float acc = 0.0f;
#pragma unroll 4
    for (int k = 0; k < NVIN; ++k)
      acc += Vsrc[row * vstride + k * 3 + d] * wh[h * NVIN + k];
    VH[row * 99 + d * 33 + h] = acc;
  }
  __syncthreads();
  // vn -> A panel columns [VNCOL, VNCOL+NH)
  for (int i = lane; i < 16 * NH; i += 32) {
    int row = i / NH, h = i % NH;
    float q = 0.0f;
#pragma unroll
    for (int d = 0; d < 3; ++d) { float t = VH[row * 99 + d * 33 + h]; q += t * t; }
    A[row * 324 + VNCOL + h] = sqrtf(fmaxf(q, 1e-8f));
  }
  __syncthreads();
  gemm_wmma<KPAD, 128>(A, 324, wswP, wsb, SOb, 128, lane);
  __syncthreads();
  // gate input (sigmoid(so) for g0/g1, raw so for g2) -> A cols 0..127
  for (int i = lane; i < 16 * 128; i += 32) {
    int row = i >> 7, j = i & 127;
    float x = SOb[i];
    A[row * 324 + j] = GSIG ? sigm(x) : x;
  }
  __syncthreads();
  gemm_wmma<128, 16>(A, 324, wsvP, wsvb, GT, 16, lane);
  __syncthreads();
  // vo[row][n][d] = (sum_h vh[row][d][h]*wv[n][h]) * sigmoid(gate[row][n])
  for (int i = lane; i < 16 * 48; i += 32) {
    int row = i / 48, j = i % 48;
    int n = j / 3, d = j % 3;
    float acc = 0.0f;
#pragma unroll 4
    for (int h = 0; h < NH; ++h)
      acc += VH[row * 99 + d * 33 + h] * wv[n * NH + h];
    VOb[row * 48 + n * 3 + d] = acc * sigm(GT[row * 16 + n]);
  }
  __syncthreads();
}

__global__ void __launch_bounds__(32) gvp_msg_kernel(
    const float* __restrict__ s, const float* __restrict__ v,
    const float* __restrict__ es, const float* __restrict__ ev,
    const int* __restrict__ ei,
    const float* wh0, const float* wswP0, const float* wsb0,
    const float* wv0, const float* wsvP0, const float* wsvb0,
    const float* wh1, const float* wswP1, const float* wsb1,
    const float* wv1, const float* wsvP1, const float* wsvb1,
    const float* wh2, const float* wswP2, const float* wsb2,
    const float* wv2, const float* wsvP2, const float* wsvb2,
    float* __restrict__ agg)
{
  __shared__ float A[16 * 324];    // activation panel (K padded to 324)
  __shared__ float MV[16 * 99];    // concatenated vector input [33][3]
  __shared__ float VH[16 * 99];    // vh [3][<=33]
  __shared__ float SOb[16 * 128];  // scalar output
  __shared__ float GT[16 * 16];    // gate
  __shared__ float VOb[16 * 48];   // vector output [16][3]

  const int lane = threadIdx.x;
  const int r0 = blockIdx.x * 16;  // 16 edge-channel rows per wave

  // ---- stage 0: gather concatenated message inputs ----
  for (int i = lane; i < 16 * 99; i += 32) {
    int row = i / 99, j = i % 99;
    int gr = r0 + row, e = gr / C_, c = gr % C_;
    int src = ei[e], dst = ei[E_ + e];
    int k = j / 3, d = j % 3;
    float val;
    if (k < VI_)       val = v[((src * C_ + c) * VI_ + k) * 3 + d];
    else if (k == VI_) val = ev[(e * C_ + c) * 3 + d];
    else               val = v[((dst * C_ + c) * VI_ + (k - VI_ - 1)) * 3 + d];
    MV[row * 99 + j] = val;
  }
  for (int i = lane; i < 16 * 288; i += 32) {
    int row = i / 288, j = i % 288;
    int gr = r0 + row, e = gr / C_, c = gr % C_;
    int src = ei[e], dst = ei[E_ + e];
    float val;
    if (j < SI_)             val = s[(src * C_ + c) * SI_ + j];
    else if (j < SI_ + SE_)  val = es[(e * C_ + c) * SE_ + (j - SI_)];
    else                     val = s[(dst * C_ + c) * SI_ + (j - SI_ - SE_)];
    A[row * 324 + j] = val;
  }
  for (int i = lane; i < 16 * 3; i += 32)      // zero K-pad cols 321..323
    A[(i / 3) * 324 + 321 + (i % 3)] = 0.0f;
  __syncthreads();

  // ---- GVP g0: nv_in=33, nh=33, K=321 (pad 324), sigmoid gate ----
  gvp_tail<33, 33, 288, 324, true>(A, VH, SOb, GT, VOb, MV, 99,
                                   wh0, wswP0, wsb0, wv0, wsvP0, wsvb0, lane);
  // silu(so0) -> A cols 0..127 for g1 scalar prefix
  for (int i = lane; i < 16 * 128; i += 32) {
    int row = i >> 7, j = i & 127;
    float x = SOb[i];
    A[row * 324 + j] = x * sigm(x);
  }
  __syncthreads();

  // ---- GVP g1: nv_in=16, nh=16, K=144, sigmoid gate ----
  gvp_tail<16, 16, 128, 144, true>(A, VH, SOb, GT, VOb, VOb, 48,
                                   wh1, wswP1, wsb1, wv1, wsvP1, wsvb1, lane);
  for (int i = lane; i < 16 * 128; i += 32) {
    int row = i >> 7, j = i & 127;
    float x = SOb[i];
    A[row * 324 + j] = x * sigm(x);
  }
  __syncthreads();

  // ---- GVP g2: no activations (gate input = raw so) ----
  gvp_tail<16, 16, 128, 144, false>(A, VH, SOb, GT, VOb, VOb, 48,
                                    wh2, wswP2, wsb2, wv2, wsvP2, wsvb2, lane);

  // ---- scatter into segment-sum accumulator agg[N][880] ----
  for (int i = lane; i < 16 * 128; i += 32) {
    int row = i >> 7, j = i & 127;
    int gr = r0 + row, e = gr / C_, c = gr % C_;
    int dst = ei[E_ + e];
    atomicAdd(&agg[dst * 880 + c * 128 + j], SOb[i]);
  }
  for (int i = lane; i < 16 * 48; i += 32) {
    int row = i / 48, j = i % 48;
    int gr = r0 + row, e = gr / C_, c = gr % C_;
    int dst = ei[E_ + e];
    atomicAdd(&agg[dst * 880 + 640 + c * 48 + j], VOb[i]);
  }
}

__global__ void zero_kernel(float* __restrict__ p, int n) {
  int i = blockIdx.x * blockDim.x + threadIdx.x;
  if (i < n) p[i] = 0.0f;
}

__global__ void cnt_kernel(const int* __restrict__ ei, float* __restrict__ cnt) {
  int e = blockIdx.x * blockDim.x + threadIdx.x;
  if (e < E_) atomicAdd(&cnt[ei[E_ + e]], 1.0f);
}

// residual + LayerNorm (scalars) + vector-norm normalization; 1 block per (n,c)
__global__ void __launch_bounds__(128) finalize_kernel(
    const float* __restrict__ s, const float* __restrict__ v,
    const float* __restrict__ agg, const float* __restrict__ cnt,
    const float* __restrict__ ln_w, const float* __restrict__ ln_b,
    float* __restrict__ ns, float* __restrict__ nv)
{
  const int nc = blockIdx.x;
  const int n = nc / C_, c = nc % C_;
  const int t = threadIdx.x;
  __shared__ float red[128];
  __shared__ float rvsh[48];
  __shared__ float sden;

  const float icnt = 1.0f / fmaxf(cnt[n], 1.0f);
  float rs = s[nc * SI_ + t] + agg[n * 880 + c * 128 + t] * icnt;
  red[t] = rs;
  __syncthreads();
  for (int off = 64; off > 0; off >>= 1) {
    if (t < off) red[t] += red[t + off];
    __syncthreads();
  }
  float mu = red[0] * (1.0f / 128.0f);
  __syncthreads();
  float dv = rs - mu;
  red[t] = dv * dv;
  __syncthreads();
  for (int off = 64; off > 0; off >>= 1) {
    if (t < off) red[t] += red[t + off];
    __syncthreads();
  }
  float var = red[0] * (1.0f / 128.0f);
  ns[nc * SO_ + t] = dv * rsqrtf(var + 1e-5f) * ln_w[t] + ln_b[t];

  if (t < 48)
    rvsh[t] = v[nc * 48 + t] + agg[n * 880 + 640 + c * 48 + t] * icnt;
  __syncthreads();
  if (t == 0) {
    float acc = 0.0f;
    for (int k = 0; k < VO_; ++k) {
      float q = 0.0f;
      for (int d = 0; d < 3; ++d) { float x = rvsh[k * 3 + d]; q += x * x; }
      acc += fmaxf(q, 1e-8f);
    }
    sden = rsqrtf(acc * (1.0f / 16.0f) + 1e-8f);
  }
  __syncthreads();
  if (t < 48) nv[nc * 48 + t] = rvsh[t] * sden;
}

// Graphormer-style shortest-path edge bias: bias[h][i][j]
__global__ void __launch_bounds__(256) bias_kernel(
    const float* __restrict__ ef, const float* __restrict__ eemb,
    const int* __restrict__ spe, float* __restrict__ bias)
{
  __shared__ float W[L_ * 128];  // edge_emb rows 1..5, each H*D = 128
  for (int i = threadIdx.x; i < L_ * 128; i += blockDim.x) W[i] = eemb[128 + i];
  __syncthreads();
  const int ij = blockIdx.x * blockDim.x + threadIdx.x;  // (i,j) pair
  float acc[H_] = {};
  int nvalid = 0;
  for (int l = 0; l < L_; ++l) {
    int idx = spe[ij * L_ + l];
    if (idx != -1) {
      ++nvalid;
      const float* row = &ef[idx * D_];
#pragma unroll
      for (int d = 0; d < D_; ++d) {
        float pv = row[d];
#pragma unroll
        for (int h = 0; h < H_; ++h) acc[h] += pv * W[l * 128 + h * 16 + d];
      }
    }
  }
  const float inv = 1.0f / (float)(nvalid < 1 ? 1 : nvalid);
#pragma unroll
  for (int h = 0; h < H_; ++h) bias[h * (N_ * N_) + ij] = acc[h] * inv;
}

extern "C" void kernel_launch(void* const* d_in, const int* in_sizes, int n_in,
                              void* d_out, int out_size, void* d_ws, size_t ws_size,
                              hipStream_t stream)
{
  (void)in_sizes; (void)n_in; (void)out_size; (void)ws_size;
  const float* s    = (const float*)d_in[0];
  const float* v    = (const float*)d_in[1];
  const float* es   = (const float*)d_in[2];
  const float* ev   = (const float*)d_in[3];
  const float* ef   = (const float*)d_in[4];
  const float* wh0  = (const float*)d_in[5];
  const float* wsw0 = (const float*)d_in[6];
  const float* wsb0 = (const float*)d_in[7];
  const float* wv0  = (const float*)d_in[8];
  const float* wsv0 = (const float*)d_in[9];
  const float* wsvb0= (const float*)d_in[10];
  const float* wh1  = (const float*)d_in[11];
  const float* wsw1 = (const float*)d_in[12];
  const float* wsb1 = (const float*)d_in[13];
  const float* wv1  = (const float*)d_in[14];
  const float* wsv1 = (const float*)d_in[15];
  const float* wsvb1= (const float*)d_in[16];
  const float* wh2  = (const float*)d_in[17];
  const float* wsw2 = (const float*)d_in[18];
  const float* wsb2 = (const float*)d_in[19];
  const float* wv2  = (const float*)d_in[20];
  const float* wsv2 = (const float*)d_in[21];
  const float* wsvb2= (const float*)d_in[22];
  const float* ln_w = (const float*)d_in[23];
  const float* ln_b = (const float*)d_in[24];
  const float* eemb = (const float*)d_in[25];
  const int*   ei   = (const int*)d_in[26];
  const int*   spe  = (const int*)d_in[27];

  float* agg  = (float*)d_ws;             // [N][880]
  float* cnt  = agg + N_ * 880;           // [N]
  float* wp0  = cnt + N_;                 // [324*128] packed g0 ws_w
  float* wp1  = wp0 + 324 * 128;          // [144*128]
  float* wp2  = wp1 + 144 * 128;          // [144*128]
  float* wpg0 = wp2 + 144 * 128;          // [128*16] packed g0 wsv_w
  float* wpg1 = wpg0 + 128 * 16;
  float* wpg2 = wpg1 + 128 * 16;

  float* ns   = (float*)d_out;            // [N,C,128]
  float* nv   = ns + N_ * C_ * SO_;       // [N,C,16,3]
  float* bias = ns + N_ * C_ * (SO_ + VO_ * 3);  // [H,N,N]

  const int nz = N_ * 880 + N_;
  zero_kernel<<<(nz + 255) / 256, 256, 0, stream>>>(agg, nz);
  cnt_kernel<<<(E_ + 255) / 256, 256, 0, stream>>>(ei, cnt);

  pack_kernel<<<(324 * 128 + 255) / 256, 256, 0, stream>>>(wsw0, wp0, 321, 324, 128);
  pack_kernel<<<(144 * 128 + 255) / 256, 256, 0, stream>>>(wsw1, wp1, 144, 144, 128);
  pack_kernel<<<(144 * 128 + 255) / 256, 256, 0, stream>>>(wsw2, wp2, 144, 144, 128);
  pack_kernel<<<(128 * 16 + 255) / 256, 256, 0, stream>>>(wsv0, wpg0, 128, 128, 16);
  pack_kernel<<<(128 * 16 + 255) / 256, 256, 0, stream>>>(wsv1, wpg1, 128, 128, 16);
  pack_kernel<<<(128 * 16 + 255) / 256, 256, 0, stream>>>(wsv2, wpg2, 128, 128, 16);

  gvp_msg_kernel<<<(E_ * C_) / 16, 32, 0, stream>>>(
      s, v, es, ev, ei,
      wh0, wp0, wsb0, wv0, wpg0, wsvb0,
      wh1, wp1, wsb1, wv1, wpg1, wsvb1,
      wh2, wp2, wsb2, wv2, wpg2, wsvb2, agg);
  finalize_kernel<<<N_ * C_, 128, 0, stream>>>(s, v, agg, cnt, ln_w, ln_b, ns, nv);
  bias_kernel<<<(N_ * N_) / 256, 256, 0, stream>>>(ef, eemb, spe, bias);
}